// Predictor_54623394071035
// MI455X (gfx1250) — compile-verified
//
#include <hip/hip_runtime.h>
#include <hip/hip_bf16.h>
#include <math.h>

typedef float v2f __attribute__((ext_vector_type(2)));
typedef float v8f __attribute__((ext_vector_type(8)));

// ---------------------------------------------------------------------------
// fp32 WMMA GEMM:  C[M,Nout] = act( A[M,K] @ W[Nout,K]^T + bias + addend )
// One wave computes a 16 x (16*NT) strip with NT independent accumulators:
// per K-step -> 1 A-fragment load, NT B-fragment loads, NT independent WMMAs
// (hides the WMMA->WMMA D-dependency latency, 4x A reuse for NT=4).
// M multiple of 16; Nout multiple of 16*NT; K multiple of 4.
// ---------------------------------------------------------------------------
template <int NT>
__global__ __launch_bounds__(32) void gemm_wmma_f32(
    const float* __restrict__ A, const float* __restrict__ W,
    const float* __restrict__ bias, const float* __restrict__ addend,
    float* __restrict__ C, int M, int K, int Nout, int act)
{
    const int m0   = blockIdx.x * 16;
    const int n0   = blockIdx.y * 16 * NT;
    const int lane = threadIdx.x;
    const int half = lane >> 4;     // 0: K pair {0,1}, 1: K pair {2,3}
    const int l15  = lane & 15;

    const float* __restrict__ arow = A + (size_t)(m0 + l15) * K;
    const float* wrow[NT];
#pragma unroll
    for (int t = 0; t < NT; ++t)
        wrow[t] = W + (size_t)(n0 + t * 16 + l15) * K;

    v8f acc[NT];
#pragma unroll
    for (int t = 0; t < NT; ++t) acc[t] = (v8f){0.f,0.f,0.f,0.f,0.f,0.f,0.f,0.f};

    for (int k = 0; k < K; k += 4) {
        const int ka = k + 2 * half;
        v2f a; a.x = arow[ka]; a.y = arow[ka + 1];
#pragma unroll
        for (int t = 0; t < NT; ++t) {
            v2f b; b.x = wrow[t][ka]; b.y = wrow[t][ka + 1];
            acc[t] = __builtin_amdgcn_wmma_f32_16x16x4_f32(
                /*neg_a=*/false, a, /*neg_b=*/false, b,
                /*c_mod=*/(short)0, acc[t], /*reuse_a=*/false, /*reuse_b=*/false);
        }
    }

#pragma unroll
    for (int t = 0; t < NT; ++t) {
        const int colc = n0 + t * 16 + l15;
        const float bv = bias ? bias[colc] : 0.0f;
#pragma unroll
        for (int j = 0; j < 8; ++j) {
            const int r = m0 + j + 8 * half;
            float v = acc[t][j] + bv;
            if (addend) v += addend[(size_t)r * Nout + colc];
            if (act) v = v > 0.f ? v : 0.f;
            C[(size_t)r * Nout + colc] = v;
        }
    }
}

// ---------------------------------------------------------------------------
// Small helpers
// ---------------------------------------------------------------------------
__device__ __forceinline__ void atomicMaxF(float* addr, float v) {
    if (v >= 0.f) atomicMax((int*)addr, __float_as_int(v));
    else          atomicMin((unsigned int*)addr, __float_as_uint(v));
}

__global__ void init_smax(float* smax) { if (threadIdx.x == 0) *smax = -__builtin_inff(); }

__global__ void concat_wrc(const float* __restrict__ wr, const float* __restrict__ wc,
                           float* __restrict__ wrc) {
    int t = blockIdx.x * blockDim.x + threadIdx.x;   // 2048 total
    if (t < 1024)       wrc[t] = wr[t];
    else if (t < 2048)  wrc[t] = wc[t - 1024];
}

// CSR row_ptr from sorted row[] (row_ptr[v] = first edge with row >= v)
__global__ void build_row_ptr(const int* __restrict__ row, int* __restrict__ rp,
                              int E, int Nn) {
    int t = blockIdx.x * blockDim.x + threadIdx.x;
    if (t > E) return;
    if (t == 0) {
        int b = row[0];
        for (int v = 0; v <= b; ++v) rp[v] = 0;
    } else if (t == E) {
        int a = row[E - 1];
        for (int v = a + 1; v <= Nn; ++v) rp[v] = E;
    } else {
        int a = row[t - 1], b = row[t];
        for (int v = a + 1; v <= b; ++v) rp[v] = t;
    }
}

// s_e = leakyrelu( dot(ar[row], ac[col]) / sqrt(8) ); store raw; wave-reduce max -> smax
__global__ __launch_bounds__(256) void edge_score(
    const float* __restrict__ arac, const int* __restrict__ row,
    const int* __restrict__ col, float* __restrict__ sexp,
    float* __restrict__ smax, int E)
{
    int e = blockIdx.x * blockDim.x + threadIdx.x;
    float s = -__builtin_inff();
    if (e < E) {
        const float* ar = arac + (size_t)row[e] * 16;      // wr half
        const float* ac = arac + (size_t)col[e] * 16 + 8;  // wc half
        float d = 0.f;
#pragma unroll
        for (int i = 0; i < 8; ++i) d += ar[i] * ac[i];
        d *= 0.35355339059327379f;                 // 1/sqrt(8)
        d = d >= 0.f ? d : 0.2f * d;               // leaky_relu
        sexp[e] = d;
        s = d;
    }
    for (int off = 16; off > 0; off >>= 1) s = fmaxf(s, __shfl_xor(s, off, 32));
    if ((threadIdx.x & 31) == 0 && s > -__builtin_inff()) atomicMaxF(smax, s);
}

__global__ void edge_exp(float* __restrict__ sexp, const float* __restrict__ smax, int E) {
    int e = blockIdx.x * blockDim.x + threadIdx.x;
    if (e < E) sexp[e] = expf(sexp[e] - *smax);
}

// One wave per node: denom = sum(sexp over its edges); agg = sum a_e * x[col_e]  (128-wide)
__global__ __launch_bounds__(256) void agg_rows(
    const float* __restrict__ x, const int* __restrict__ col,
    const float* __restrict__ sexp, const int* __restrict__ rp,
    float* __restrict__ agg, int Nn)
{
    int node = blockIdx.x * (blockDim.x >> 5) + (threadIdx.x >> 5);
    if (node >= Nn) return;
    const int lane = threadIdx.x & 31;
    const int e0 = rp[node], e1 = rp[node + 1];

    float denom = 0.f;
    for (int e = e0 + lane; e < e1; e += 32) denom += sexp[e];
    for (int off = 16; off > 0; off >>= 1) denom += __shfl_xor(denom, off, 32);
    const float inv = 1.0f / (denom + 1e-15f);

    float4 acc = {0.f, 0.f, 0.f, 0.f};
    for (int e = e0; e < e1; ++e) {
        const float a = sexp[e] * inv;
        const float4 v = ((const float4*)(x + (size_t)col[e] * 128))[lane];
        acc.x += a * v.x; acc.y += a * v.y; acc.z += a * v.z; acc.w += a * v.w;
    }
    ((float4*)(agg + (size_t)node * 128))[lane] = acc;
}

// z[b, 0:128] = h[src]*h[dst];  z[b, 128:192] = emb_h[hh]
__global__ __launch_bounds__(192) void gather_z(
    const float* __restrict__ h, const int* __restrict__ src,
    const int* __restrict__ dst, const int* __restrict__ hh,
    const float* __restrict__ emb, float* __restrict__ z)
{
    int b = blockIdx.x, t = threadIdx.x;
    float v;
    if (t < 128) v = h[(size_t)src[b] * 128 + t] * h[(size_t)dst[b] * 128 + t];
    else         v = emb[(size_t)hh[b] * 64 + (t - 128)];
    z[(size_t)b * 192 + t] = v;
}

// LayerNorm(192) + ReLU, one wave per row
__global__ __launch_bounds__(256) void ln_relu(
    const float* __restrict__ in, float* __restrict__ out,
    const float* __restrict__ g, const float* __restrict__ bta, int M)
{
    int rowi = blockIdx.x * (blockDim.x >> 5) + (threadIdx.x >> 5);
    if (rowi >= M) return;
    const int lane = threadIdx.x & 31;
    const float* r = in + (size_t)rowi * 192;
    float vals[6], s = 0.f, s2 = 0.f;
#pragma unroll
    for (int i = 0; i < 6; ++i) {
        float v = r[lane + 32 * i];
        vals[i] = v; s += v; s2 += v * v;
    }
    for (int off = 16; off > 0; off >>= 1) { s += __shfl_xor(s, off, 32); s2 += __shfl_xor(s2, off, 32); }
    const float mu  = s * (1.0f / 192.0f);
    const float var = s2 * (1.0f / 192.0f) - mu * mu;
    const float inv = rsqrtf(var + 1e-5f);
    float* o = out + (size_t)rowi * 192;
#pragma unroll
    for (int i = 0; i < 6; ++i) {
        const int c = lane + 32 * i;
        float v = (vals[i] - mu) * inv * g[c] + bta[c];
        o[c] = v > 0.f ? v : 0.f;
    }
}

// out[b] = sigmoid( dot(z64[b], f1_w) + f1_b )
__global__ void final_dot(const float* __restrict__ z64, const float* __restrict__ w,
                          const float* __restrict__ b1, float* __restrict__ out, int B)
{
    int i = blockIdx.x * blockDim.x + threadIdx.x;
    if (i >= B) return;
    float s = b1[0];
    const float* r = z64 + (size_t)i * 64;
#pragma unroll 8
    for (int k = 0; k < 64; ++k) s += r[k] * w[k];
    out[i] = 1.0f / (1.0f + expf(-s));
}

// ---------------------------------------------------------------------------
// Host orchestration
// ---------------------------------------------------------------------------
static void run_gtm_layer(const float* X,
                          const float* wres, const float* bres,
                          const float* wr, const float* wc,
                          const float* wx, const float* bx,
                          const float* wo, const float* bo,
                          const int* row, const int* col, const int* row_ptr,
                          float* wrc, float* arac, float* sexp, float* smax,
                          float* R, float* AGG, float* Y, float* H,
                          int Nn, int E, int relu_out, hipStream_t stream)
{
    const int mt = Nn / 16;
    concat_wrc<<<8, 256, 0, stream>>>(wr, wc, wrc);
    // arac = X @ wrc^T                            [Nn,16]
    gemm_wmma_f32<1><<<dim3(mt, 1), 32, 0, stream>>>(X, wrc, nullptr, nullptr, arac, Nn, 128, 16, 0);
    // R = X @ wres^T + bres                       [Nn,128]
    gemm_wmma_f32<4><<<dim3(mt, 2), 32, 0, stream>>>(X, wres, bres, nullptr, R, Nn, 128, 128, 0);
    // edge softmax numerators
    init_smax<<<1, 32, 0, stream>>>(smax);
    edge_score<<<(E + 255) / 256, 256, 0, stream>>>(arac, row, col, sexp, smax, E);
    edge_exp<<<(E + 255) / 256, 256, 0, stream>>>(sexp, smax, E);
    // per-row softmax normalize + weighted gather  [Nn,128]
    agg_rows<<<(Nn + 7) / 8, 256, 0, stream>>>(X, col, sexp, row_ptr, AGG, Nn);
    // Y = AGG @ wx^T + bx + R
    gemm_wmma_f32<4><<<dim3(mt, 2), 32, 0, stream>>>(AGG, wx, bx, R, Y, Nn, 128, 128, 0);
    // H = act( Y @ wo^T + bo )
    gemm_wmma_f32<4><<<dim3(mt, 2), 32, 0, stream>>>(Y, wo, bo, nullptr, H, Nn, 128, 128, relu_out);
}

extern "C" void kernel_launch(void* const* d_in, const int* in_sizes, int n_in,
                              void* d_out, int out_size, void* d_ws, size_t ws_size,
                              hipStream_t stream)
{
    // ---- inputs (setup_inputs dict order) ----
    const float* x      = (const float*)d_in[0];
    const int*   row    = (const int*)  d_in[1];
    const int*   col    = (const int*)  d_in[2];
    const int*   eb_src = (const int*)  d_in[3];
    const int*   eb_dst = (const int*)  d_in[4];
    const int*   eb_h   = (const int*)  d_in[5];
    const float* l_wres[2] = { (const float*)d_in[6],  (const float*)d_in[14] };
    const float* l_bres[2] = { (const float*)d_in[7],  (const float*)d_in[15] };
    const float* l_wr[2]   = { (const float*)d_in[8],  (const float*)d_in[16] };
    const float* l_wc[2]   = { (const float*)d_in[9],  (const float*)d_in[17] };
    const float* l_wx[2]   = { (const float*)d_in[10], (const float*)d_in[18] };
    const float* l_bx[2]   = { (const float*)d_in[11], (const float*)d_in[19] };
    const float* l_wo[2]   = { (const float*)d_in[12], (const float*)d_in[20] };
    const float* l_bo[2]   = { (const float*)d_in[13], (const float*)d_in[21] };
    const float* emb_h  = (const float*)d_in[22];
    const float* mlp0_w = (const float*)d_in[23];
    const float* mlp0_b = (const float*)d_in[24];
    const float* mlp1_w = (const float*)d_in[25];
    const float* mlp1_b = (const float*)d_in[26];
    const float* ln_g   = (const float*)d_in[27];
    const float* ln_b   = (const float*)d_in[28];
    const float* f0_w   = (const float*)d_in[29];
    const float* f0_b   = (const float*)d_in[30];
    const float* f1_w   = (const float*)d_in[31];
    const float* f1_b   = (const float*)d_in[32];
    float* out = (float*)d_out;

    const int Nn = in_sizes[0] / 128;   // 100000 (multiple of 16)
    const int E  = in_sizes[1];         // 800000
    const int B  = in_sizes[3];         // 65536  (multiple of 16)

    // ---- workspace layout ----
    char* ws = (char*)d_ws;
    size_t off = 0;
    auto alloc = [&](size_t bytes) -> char* {
        char* p = ws + off;
        off += (bytes + 255) & ~(size_t)255;
        return p;
    };
    float* NB0  = (float*)alloc((size_t)Nn * 128 * 4);
    float* NB1  = (float*)alloc((size_t)Nn * 128 * 4);
    float* NB2  = (float*)alloc((size_t)Nn * 128 * 4);
    float* NB3  = (float*)alloc((size_t)Nn * 128 * 4);
    float* arac = (float*)alloc((size_t)Nn * 16 * 4);
    float* sexp = (float*)alloc((size_t)E * 4);
    int*   rp   = (int*)  alloc((size_t)(Nn + 1) * 4);
    float* wrc  = (float*)alloc(16 * 128 * 4);
    float* smax = (float*)alloc(256);
    (void)ws_size; (void)n_in; (void)out_size;

    // CSR row pointers (row is sorted); same for both layers
    build_row_ptr<<<(E + 1 + 255) / 256, 256, 0, stream>>>(row, rp, E, Nn);

    // ---- GNN layers ----
    // layer 0: X=x  -> H0=NB3 (with ReLU)
    run_gtm_layer(x, l_wres[0], l_bres[0], l_wr[0], l_wc[0], l_wx[0], l_bx[0],
                  l_wo[0], l_bo[0], row, col, rp, wrc, arac, sexp, smax,
                  NB0, NB1, NB2, NB3, Nn, E, /*relu=*/1, stream);
    // layer 1: X=NB3 -> H1=NB0 (no activation)
    run_gtm_layer(NB3, l_wres[1], l_bres[1], l_wr[1], l_wc[1], l_wx[1], l_bx[1],
                  l_wo[1], l_bo[1], row, col, rp, wrc, arac, sexp, smax,
                  NB1, NB2, NB3, NB0, Nn, E, /*relu=*/0, stream);
    float* H1 = NB0;

    // ---- edge-batch MLP head ----
    const int bt = B / 16;
    float* z0 = NB2;  // [B,192]
    gather_z<<<B, 192, 0, stream>>>(H1, eb_src, eb_dst, eb_h, emb_h, z0);

    float* t0 = NB1;  // pre-LN
    float* z1 = NB3;
    gemm_wmma_f32<4><<<dim3(bt, 3), 32, 0, stream>>>(z0, mlp0_w, mlp0_b, nullptr, t0, B, 192, 192, 0);
    ln_relu<<<(B + 7) / 8, 256, 0, stream>>>(t0, z1, ln_g, ln_b, B);

    float* z2 = NB2;
    gemm_wmma_f32<4><<<dim3(bt, 3), 32, 0, stream>>>(z1, mlp1_w, mlp1_b, nullptr, t0, B, 192, 192, 0);
    ln_relu<<<(B + 7) / 8, 256, 0, stream>>>(t0, z2, ln_g, ln_b, B);

    float* z64 = NB1; // [B,64]
    gemm_wmma_f32<4><<<dim3(bt, 1), 32, 0, stream>>>(z2, f0_w, f0_b, nullptr, z64, B, 192, 64, 0);

    final_dot<<<(B + 255) / 256, 256, 0, stream>>>(z64, f1_w, f1_b, out, B);
}